// Group_8744553414804
// MI455X (gfx1250) — compile-verified
//
#include <hip/hip_runtime.h>

#define N_PTS 8192
#define B_BATCH 8
#define T_STEPS 4
#define N_CENTER_ALL 416   // 128 + (32+64)*3
#define NUM_GROUP 128
#define GROUP_SIZE 32
#define FLT_MAX_C 3.402823466e+38f

typedef __attribute__((ext_vector_type(2))) float v2f;
typedef __attribute__((ext_vector_type(8))) float v8f;

__device__ __forceinline__ void argmaxCombine(float& d, int& i, float od, int oi) {
    if (od > d || (od == d && oi < i)) { d = od; i = oi; }
}
__device__ __forceinline__ void argminCombine(float& d, int& i, float od, int oi) {
    if (od < d || (od == d && oi < i)) { d = od; i = oi; }
}

// ---------------------------------------------------------------------------
// Kernel 1: furthest point sampling. One workgroup per batch (8 WGs).
// 1024 threads; 8 points/thread held in registers; per-round argmax via
// wave32 shuffle butterfly + cross-wave LDS reduce. Sequential 415 rounds.
// ---------------------------------------------------------------------------
__global__ void __launch_bounds__(1024) fps_kernel(const float* __restrict__ xyz,
                                                   float* __restrict__ center_all) {
    const int b   = blockIdx.x;
    const int tid = threadIdx.x;
    const float* pts = xyz + (size_t)b * N_PTS * 3;

    float px[8], py[8], pz[8], dist[8];
#pragma unroll
    for (int k = 0; k < 8; ++k) {
        const int j = k * 1024 + tid;
        px[k] = pts[3 * j + 0];
        py[k] = pts[3 * j + 1];
        pz[k] = pts[3 * j + 2];
        dist[k] = 1e10f;   // matches reference init
    }

    __shared__ float s_px, s_py, s_pz;
    __shared__ float s_rd[32];
    __shared__ int   s_ri[32];

    if (tid == 0) {
        // idx[0] = 0 (reference starts at point 0)
        s_px = px[0]; s_py = py[0]; s_pz = pz[0];
        center_all[((size_t)b * N_CENTER_ALL + 0) * 3 + 0] = px[0];
        center_all[((size_t)b * N_CENTER_ALL + 0) * 3 + 1] = py[0];
        center_all[((size_t)b * N_CENTER_ALL + 0) * 3 + 2] = pz[0];
    }

    for (int it = 1; it < N_CENTER_ALL; ++it) {
        __syncthreads();                       // s_p* visible; s_rd free
        const float qx = s_px, qy = s_py, qz = s_pz;

        float bd = -1.0f; int bi = 0x7fffffff;
#pragma unroll
        for (int k = 0; k < 8; ++k) {
            const float dx = px[k] - qx, dy = py[k] - qy, dz = pz[k] - qz;
            const float d  = dx * dx + dy * dy + dz * dz;
            dist[k] = fminf(dist[k], d);
            const int j = k * 1024 + tid;
            if (dist[k] > bd || (dist[k] == bd && j < bi)) { bd = dist[k]; bi = j; }
        }
        // wave32 butterfly argmax
#pragma unroll
        for (int m = 16; m >= 1; m >>= 1) {
            const float od = __shfl_xor(bd, m, 32);
            const int   oi = __shfl_xor(bi, m, 32);
            argmaxCombine(bd, bi, od, oi);
        }
        if ((tid & 31) == 0) { s_rd[tid >> 5] = bd; s_ri[tid >> 5] = bi; }
        __syncthreads();
        if (tid == 0) {
            float fd = s_rd[0]; int fi = s_ri[0];
            for (int w = 1; w < 32; ++w) argmaxCombine(fd, fi, s_rd[w], s_ri[w]);
            const float nx = pts[3 * fi + 0];
            const float ny = pts[3 * fi + 1];
            const float nz = pts[3 * fi + 2];
            s_px = nx; s_py = ny; s_pz = nz;
            center_all[((size_t)b * N_CENTER_ALL + it) * 3 + 0] = nx;
            center_all[((size_t)b * N_CENTER_ALL + it) * 3 + 1] = ny;
            center_all[((size_t)b * N_CENTER_ALL + it) * 3 + 2] = nz;
        }
    }
}

// ---------------------------------------------------------------------------
// Kernel 2: slice center_all into per-timestep 128 centers; also emits the
// `center` output. 32 blocks (tb) x 128 threads (group id).
// ---------------------------------------------------------------------------
__global__ void centers_kernel(const float* __restrict__ center_all,
                               float* __restrict__ centers,
                               float* __restrict__ out_center) {
    const int tb = blockIdx.x;
    const int g  = threadIdx.x;                // 0..127
    const int t  = tb >> 3;
    const int b  = tb & 7;
    // c1: center_all[t*32 + g]          for g in [0,64)
    // c2: center_all[(t-1)*64 + 224 + (g-64)] = t*64 + 96 + g  for g in [64,128)
    const int src = (g < 64) ? (t * 32 + g) : (t * 64 + 96 + g);
#pragma unroll
    for (int c = 0; c < 3; ++c) {
        const float v = center_all[((size_t)b * N_CENTER_ALL + src) * 3 + c];
        centers[((size_t)tb * NUM_GROUP + g) * 3 + c]    = v;
        out_center[((size_t)tb * NUM_GROUP + g) * 3 + c] = v;
    }
}

// ---------------------------------------------------------------------------
// Kernel 3: WMMA distance scores.  score = r^2 - 2 q.r  (q^2 dropped: order-
// preserving per query).  One wave per WG; grid (ntile16-block=16, mtile=8,
// tb-in-pass).  A(16x4 f32): row m = (-2qx,-2qy,-2qz, 1);  B(4x16 f32):
// col n = (rx, ry, rz, r^2).  One V_WMMA_F32_16X16X4_F32 per 16x16 tile.
// Layouts per ISA 7.12.2: A lanes 0-15 carry M rows, VGPR0=K0|K2, VGPR1=K1|K3
// by half-wave; C VGPR v holds rows M=v (lanes 0-15) and M=v+8 (lanes 16-31).
// ---------------------------------------------------------------------------
__global__ void __launch_bounds__(32) score_kernel(const float* __restrict__ xyz,
                                                   const float* __restrict__ centers,
                                                   float* __restrict__ scores,
                                                   int tb0) {
    const int lane  = threadIdx.x;
    const int n     = lane & 15;
    const int half  = lane >> 4;
    const int tbl   = blockIdx.z;          // local tb within this pass
    const int tb    = tb0 + tbl;
    const int qbase = blockIdx.y * 16;
    const int rbase = blockIdx.x * 512;

    // A matrix fragment (constant over the whole ref loop)
    const float* qp = centers + ((size_t)tb * NUM_GROUP + qbase + n) * 3;
    const float qx = qp[0], qy = qp[1], qz = qp[2];
    v2f a;
    a.x = half ? (-2.0f * qz) : (-2.0f * qx);
    a.y = half ? 1.0f         : (-2.0f * qy);

    const float* rpts = xyz + (size_t)tb * N_PTS * 3;
    float* srow = scores + ((size_t)tbl * NUM_GROUP + qbase) * N_PTS;

    for (int j = 0; j < 32; ++j) {
        const int ref = rbase + j * 16 + n;
        const float* rp = rpts + (size_t)ref * 3;
        const float rx = rp[0], ry = rp[1], rz = rp[2];
        const float r2 = rx * rx + ry * ry + rz * rz;
        v2f bb;
        bb.x = half ? rz : rx;
        bb.y = half ? r2 : ry;

        v8f c = {};
        // 8 args: (neg_a, A, neg_b, B, c_mod, C, reuse_a, reuse_b)
        c = __builtin_amdgcn_wmma_f32_16x16x4_f32(
                false, a, false, bb, (short)0, c, false, false);

#pragma unroll
        for (int v = 0; v < 8; ++v) {
            const int m = v + 8 * half;           // C layout: VGPR v -> M=v / v+8
            srow[(size_t)m * N_PTS + ref] = c[v];
        }
    }
}

// ---------------------------------------------------------------------------
// Kernel 4: top-32 selection per query + gather + center-subtract.
// One WG (256 thr) per (query, tb).  Score row staged in 32KB LDS; 32 rounds
// of masked argmin (tie -> lowest index, matching jax.lax.top_k order).
// ---------------------------------------------------------------------------
__global__ void __launch_bounds__(256) topk_kernel(const float* __restrict__ xyz,
                                                   const float* __restrict__ centers,
                                                   const float* __restrict__ scores,
                                                   float* __restrict__ out_nb,
                                                   int tb0) {
    __shared__ float sd[N_PTS];     // 32 KB of 320 KB LDS
    __shared__ float s_rd[8];
    __shared__ int   s_ri[8];
    __shared__ int   sel[GROUP_SIZE];

    const int q   = blockIdx.x;
    const int tbl = blockIdx.y;
    const int tb  = tb0 + tbl;
    const int tid = threadIdx.x;

    const float* srow = scores + ((size_t)tbl * NUM_GROUP + q) * N_PTS;
    for (int k = tid; k < N_PTS; k += 256) sd[k] = srow[k];
    __syncthreads();

    for (int r = 0; r < GROUP_SIZE; ++r) {
        float bd = FLT_MAX_C; int bi = 0x7fffffff;
#pragma unroll 4
        for (int k = 0; k < N_PTS / 256; ++k) {
            const int j = k * 256 + tid;
            const float d = sd[j];
            if (d < bd || (d == bd && j < bi)) { bd = d; bi = j; }
        }
#pragma unroll
        for (int m = 16; m >= 1; m >>= 1) {
            const float od = __shfl_xor(bd, m, 32);
            const int   oi = __shfl_xor(bi, m, 32);
            argminCombine(bd, bi, od, oi);
        }
        if ((tid & 31) == 0) { s_rd[tid >> 5] = bd; s_ri[tid >> 5] = bi; }
        __syncthreads();
        if (tid == 0) {
            float fd = s_rd[0]; int fi = s_ri[0];
            for (int w = 1; w < 8; ++w) argminCombine(fd, fi, s_rd[w], s_ri[w]);
            sel[r]  = fi;
            sd[fi]  = FLT_MAX_C;     // mask out for next round
        }
        __syncthreads();
    }

    // gather neighbors, subtract center, write (T,B,128,32,3) region
    const float cx = centers[((size_t)tb * NUM_GROUP + q) * 3 + 0];
    const float cy = centers[((size_t)tb * NUM_GROUP + q) * 3 + 1];
    const float cz = centers[((size_t)tb * NUM_GROUP + q) * 3 + 2];
    if (tid < GROUP_SIZE * 3) {
        const int nn = tid / 3, cc = tid % 3;
        const int pidx = sel[nn];
        const float pv = xyz[((size_t)tb * N_PTS + pidx) * 3 + cc];
        const float cv = (cc == 0) ? cx : (cc == 1) ? cy : cz;
        out_nb[(((size_t)tb * NUM_GROUP + q) * GROUP_SIZE + nn) * 3 + cc] = pv - cv;
    }
}

// ---------------------------------------------------------------------------
// Host launcher.  Workspace layout:
//   [0 .. 40960)            center_all  (8 x 416 x 3 f32)
//   [40960 .. 90112)        centers     (32 x 128 x 3 f32)
//   [90112 .. )             scores      (tb_per_pass x 128 x 8192 f32)
// tb_per_pass adapts to ws_size; stream ordering serializes region reuse.
// ---------------------------------------------------------------------------
extern "C" void kernel_launch(void* const* d_in, const int* in_sizes, int n_in,
                              void* d_out, int out_size, void* d_ws, size_t ws_size,
                              hipStream_t stream) {
    (void)in_sizes; (void)n_in; (void)out_size;
    const float* xyz = (const float*)d_in[0];
    float* out_nb     = (float*)d_out;                                   // 393216 f32
    float* out_center = out_nb + (size_t)T_STEPS * B_BATCH * NUM_GROUP * GROUP_SIZE * 3;

    char*  ws         = (char*)d_ws;
    float* center_all = (float*)(ws);
    float* centers    = (float*)(ws + 40960);
    float* scores     = (float*)(ws + 90112);

    const size_t fixed = 90112;
    const size_t avail = (ws_size > fixed) ? (ws_size - fixed) : 0;
    int tbp = 32;
    while (tbp > 1 && (size_t)tbp * NUM_GROUP * N_PTS * sizeof(float) > avail) tbp >>= 1;

    fps_kernel<<<dim3(B_BATCH), dim3(1024), 0, stream>>>(xyz, center_all);
    centers_kernel<<<dim3(T_STEPS * B_BATCH), dim3(NUM_GROUP), 0, stream>>>(
        center_all, centers, out_center);

    for (int tb0 = 0; tb0 < T_STEPS * B_BATCH; tb0 += tbp) {
        const int remaining = T_STEPS * B_BATCH - tb0;
        const int cnt = remaining < tbp ? remaining : tbp;
        score_kernel<<<dim3(16, 8, cnt), dim3(32), 0, stream>>>(xyz, centers, scores, tb0);
        topk_kernel<<<dim3(NUM_GROUP, cnt), dim3(256), 0, stream>>>(
            xyz, centers, scores, out_nb, tb0);
    }
}